// AGNN_52673478918326
// MI455X (gfx1250) — compile-verified
//
#include <hip/hip_runtime.h>

#define N_NODES 63
#define IN_F    2048
#define HID_F   1024
#define OUT_F   300
#define OUT_FP  304      // OUT_F padded to 16
#define BATCH   2048
#define SPLIT   8

typedef __attribute__((ext_vector_type(2))) float v2f;
typedef __attribute__((ext_vector_type(8))) float v8f;

// ---------------------------------------------------------------------------
// 1) Batch mean, stage 1: partial sums over batch chunks (HBM-bound kernel).
//    labelgcn is [B, N, IN] f32; we reduce over B. float4 vectorized.
// ---------------------------------------------------------------------------
__global__ void k_mean_partial(const float4* __restrict__ in, float4* __restrict__ part) {
    const int NC4 = (N_NODES * IN_F) / 4;                  // 32256 float4 per batch slice
    int idx4 = blockIdx.x * blockDim.x + threadIdx.x;      // 0..NC4-1
    int s = blockIdx.y;                                    // split id
    float4 acc; acc.x = 0.f; acc.y = 0.f; acc.z = 0.f; acc.w = 0.f;
    const float4* p = in + (size_t)(s * (BATCH / SPLIT)) * NC4 + idx4;
    #pragma unroll 4
    for (int b = 0; b < BATCH / SPLIT; ++b) {
        float4 v = p[(size_t)b * NC4];
        acc.x += v.x; acc.y += v.y; acc.z += v.z; acc.w += v.w;
    }
    part[(size_t)s * NC4 + idx4] = acc;
}

// 2) Batch mean, stage 2: combine SPLIT partials, scale by 1/B -> lg rows 0..62
__global__ void k_mean_finish(const float4* __restrict__ part, float4* __restrict__ lg) {
    const int NC4 = (N_NODES * IN_F) / 4;
    int idx4 = blockIdx.x * blockDim.x + threadIdx.x;
    float4 acc; acc.x = 0.f; acc.y = 0.f; acc.z = 0.f; acc.w = 0.f;
    #pragma unroll
    for (int s = 0; s < SPLIT; ++s) {
        float4 v = part[(size_t)s * NC4 + idx4];
        acc.x += v.x; acc.y += v.y; acc.z += v.z; acc.w += v.w;
    }
    const float inv = 1.0f / (float)BATCH;
    acc.x *= inv; acc.y *= inv; acc.z *= inv; acc.w *= inv;
    lg[idx4] = acc;
}

// Utility: zero-fill n floats
__global__ void k_zero(float* __restrict__ p, int n) {
    int i = blockIdx.x * blockDim.x + threadIdx.x;
    if (i < n) p[i] = 0.f;
}

// Utility: dst[0..n_tot) = src padded with zeros past n_src (same row-major layout)
__global__ void k_pad(const float* __restrict__ src, float* __restrict__ dst,
                      int n_src, int n_tot) {
    int i = blockIdx.x * blockDim.x + threadIdx.x;
    if (i < n_tot) dst[i] = (i < n_src) ? src[i] : 0.f;
}

// ---------------------------------------------------------------------------
// 3) Gram matrix x = L @ L.T, L is [63, F]. One wave per (i,j) dot product.
// ---------------------------------------------------------------------------
__global__ void k_gram(const float* __restrict__ L, int F, float* __restrict__ x) {
    int i = blockIdx.x / N_NODES;
    int j = blockIdx.x % N_NODES;
    const float* a = L + (size_t)i * F;
    const float* b = L + (size_t)j * F;
    float s = 0.f;
    for (int f = threadIdx.x; f < F; f += 32) s += a[f] * b[f];
    #pragma unroll
    for (int o = 16; o > 0; o >>= 1) s += __shfl_down(s, o, 32);
    if (threadIdx.x == 0) x[blockIdx.x] = s;
}

// ---------------------------------------------------------------------------
// 4) Stats: st[0]=max(x) ("mn" in ref), st[1]=min(x) ("mx"), st[2]=trace(xn)
//    with xn = sqrt((x - max) / (min - max)).
// ---------------------------------------------------------------------------
__global__ void k_stats(const float* __restrict__ x, float* __restrict__ st) {
    __shared__ float smx[256], smn[256], str[64];
    int t = threadIdx.x;
    float mx = -3.402823466e38f, mn = 3.402823466e38f;
    for (int i = t; i < N_NODES * N_NODES; i += 256) {
        float v = x[i];
        mx = fmaxf(mx, v);
        mn = fminf(mn, v);
    }
    smx[t] = mx; smn[t] = mn;
    __syncthreads();
    for (int o = 128; o > 0; o >>= 1) {
        if (t < o) { smx[t] = fmaxf(smx[t], smx[t + o]); smn[t] = fminf(smn[t], smn[t + o]); }
        __syncthreads();
    }
    float Xmax = smx[0], Xmin = smn[0];
    float d = 0.f;
    if (t < N_NODES) {
        float v = x[t * N_NODES + t];
        d = sqrtf((v - Xmax) / (Xmin - Xmax));
    }
    if (t < 64) str[t] = d;
    __syncthreads();
    for (int o = 32; o > 0; o >>= 1) {
        if (t < o) str[t] += str[t + o];
        __syncthreads();
    }
    if (t == 0) { st[0] = Xmax; st[1] = Xmin; st[2] = str[0]; }
}

// ---------------------------------------------------------------------------
// 5) S = rownorm(adj * xn / trace(xn)); one 64-thread block per row.
// ---------------------------------------------------------------------------
__global__ void k_make_S(const float* __restrict__ x, const float* __restrict__ adj,
                         const float* __restrict__ st, float* __restrict__ S) {
    __shared__ float row[64];
    int i = blockIdx.x, j = threadIdx.x;
    float Xmax = st[0], Xmin = st[1], tr = st[2];
    float s = 0.f;
    if (j < N_NODES) {
        float v = x[i * N_NODES + j];
        float xn = sqrtf((v - Xmax) / (Xmin - Xmax));
        s = adj[i * N_NODES + j] * (xn / tr);
    }
    row[j] = s;
    __syncthreads();
    #pragma unroll
    for (int o = 32; o > 0; o >>= 1) {
        if (j < o) row[j] += row[j + o];
        __syncthreads();
    }
    float rs = row[0];
    rs = (rs == 0.f) ? 1.f : rs;
    if (j < N_NODES) S[i * N_NODES + j] = s / rs;
}

// ---------------------------------------------------------------------------
// 6) M = relu(A @ W.T + bias) via V_WMMA_F32_16X16X4_F32 (full f32 precision).
//    A: [Mpad, F] row-major zero-padded (Mpad multiple of 16, rows >= 63 zero).
//    W: [Hpad, F] row-major zero-padded (Hpad = gridDim.x*16).
//    bias: [Hpad] zero-padded. All inner-loop loads are UNCONDITIONAL b64:
//    no exec-mask predication in the hot loop.
//    A-operand layout (ISA 7.12.2, 32-bit A 16x4):
//      lane<16 : {A[m0+lane, f], A[m0+lane, f+1]}
//      lane>=16: {A[m0+lane-16, f+2], A[m0+lane-16, f+3]}
//    B = W.T slice (4x16), mirrored layout using W rows n0+(lane&15).
//    C/D layout: VGPR r -> M = r (lanes 0-15) / r+8 (lanes 16-31), N = lane&15.
// ---------------------------------------------------------------------------
__global__ void k_gemm_relu_wmma(const float* __restrict__ A, const float* __restrict__ W,
                                 const float* __restrict__ bias, int F, int H,
                                 float* __restrict__ out) {
    int lane = threadIdx.x;
    int n0 = blockIdx.x * 16;
    int m0 = blockIdx.y * 16;
    int lm = lane & 15;
    int koff = (lane >> 4) << 1;               // 0 or 2
    const float* ap = A + (size_t)(m0 + lm) * F + koff;
    const float* wp = W + (size_t)(n0 + lm) * F + koff;

    v8f c = {0.f, 0.f, 0.f, 0.f, 0.f, 0.f, 0.f, 0.f};
    #pragma unroll 8
    for (int f = 0; f < F; f += 4) {
        v2f a = *(const v2f*)(ap + f);         // global_load_b64, unconditional
        v2f b = *(const v2f*)(wp + f);         // global_load_b64, unconditional
        // 8 args: (neg_a, A, neg_b, B, c_mod, C, reuse_a, reuse_b)
        c = __builtin_amdgcn_wmma_f32_16x16x4_f32(false, a, false, b, (short)0, c,
                                                  false, false);
    }

    int col = n0 + lm;
    int rbase = m0 + ((lane >> 4) << 3);
    float bv = bias[col];                       // bias padded -> always in-bounds
    #pragma unroll
    for (int r = 0; r < 8; ++r) {
        int rowi = rbase + r;
        if (rowi < N_NODES && col < H) {
            float v = c[r] + bv;
            out[(size_t)rowi * H + col] = fmaxf(v, 0.f);
        }
    }
}

// ---------------------------------------------------------------------------
// 7) l[n,h] = sum_k S[n,k] * Mflat[h*63 + k]   (the reshape(H,63) trick:
//    inner 63 elements contiguous in M's flat storage). Block per h.
//    outH is the row stride of the (possibly padded) output buffer.
// ---------------------------------------------------------------------------
__global__ void k_smm(const float* __restrict__ S, const float* __restrict__ Mflat,
                      int outH, float* __restrict__ out) {
    __shared__ float mrow[N_NODES];
    int h = blockIdx.x, t = threadIdx.x;
    if (t < N_NODES) mrow[t] = Mflat[(size_t)h * N_NODES + t];
    __syncthreads();
    if (t < N_NODES) {
        const float* sr = S + t * N_NODES;
        float acc = 0.f;
        #pragma unroll
        for (int k = 0; k < N_NODES; ++k) acc += sr[k] * mrow[k];
        out[(size_t)t * outH + h] = acc;
    }
}

// ---------------------------------------------------------------------------
extern "C" void kernel_launch(void* const* d_in, const int* in_sizes, int n_in,
                              void* d_out, int out_size, void* d_ws, size_t ws_size,
                              hipStream_t stream) {
    (void)in_sizes; (void)n_in; (void)out_size; (void)ws_size;

    const float* labelgcn = (const float*)d_in[0];   // [2048, 63, 2048]
    const float* adj      = (const float*)d_in[1];   // [63, 63]
    const float* W1       = (const float*)d_in[2];   // [1024, 2048]
    const float* b1       = (const float*)d_in[3];   // [1024]
    const float* W2       = (const float*)d_in[4];   // [300, 1024]
    const float* b2       = (const float*)d_in[5];   // [300]

    float* out = (float*)d_out;
    float* l2  = out;                                 // [63, 300]
    float* S1  = out + OUT_F * N_NODES;               // [63, 63]
    float* S2  = S1 + N_NODES * N_NODES;              // [63, 63]

    // workspace layout (floats); padded operand buffers keep the WMMA inner
    // loop free of exec-mask predication.
    const int NC = N_NODES * IN_F;                    // 129024
    float* ws   = (float*)d_ws;
    float* part = ws;                                 // SPLIT * NC
    float* lgp  = part + (size_t)SPLIT * NC;          // [64, 2048]
    float* M1   = lgp + 64 * IN_F;                    // [63, 1024] flat
    float* l1p  = M1 + N_NODES * HID_F;               // [64, 1024]
    float* M2   = l1p + 64 * HID_F;                   // [63, 300] flat
    float* W2p  = M2 + N_NODES * OUT_F;               // [304, 1024]
    float* b2p  = W2p + OUT_FP * HID_F;               // [304]
    float* x    = b2p + OUT_FP;                       // 3969 (pad 3972)
    float* st   = x + 3972;                           // 3 floats

    const int NC4 = NC / 4;                           // 32256

    // lg = mean(labelgcn, axis=0) -> lgp rows 0..62; zero pad row 63
    k_mean_partial<<<dim3(NC4 / 256, SPLIT), 256, 0, stream>>>(
        (const float4*)labelgcn, (float4*)part);
    k_mean_finish<<<NC4 / 256, 256, 0, stream>>>((const float4*)part, (float4*)lgp);
    k_zero<<<(IN_F + 255) / 256, 256, 0, stream>>>(lgp + NC, IN_F);

    // pad W2/b2 to 304 rows (independent of the pipeline above)
    k_pad<<<(OUT_FP * HID_F + 255) / 256, 256, 0, stream>>>(W2, W2p,
        OUT_F * HID_F, OUT_FP * HID_F);
    k_pad<<<(OUT_FP + 255) / 256, 256, 0, stream>>>(b2, b2p, OUT_F, OUT_FP);

    // S1 from lg
    k_gram<<<N_NODES * N_NODES, 32, 0, stream>>>(lgp, IN_F, x);
    k_stats<<<1, 256, 0, stream>>>(x, st);
    k_make_S<<<N_NODES, 64, 0, stream>>>(x, adj, st, S1);

    // l1 = S1 @ reshape(relu(lg @ W1.T + b1), (1024, 63)).T ; pad row 63 = 0
    k_gemm_relu_wmma<<<dim3(HID_F / 16, 4), 32, 0, stream>>>(lgp, W1, b1, IN_F, HID_F, M1);
    k_smm<<<HID_F, 64, 0, stream>>>(S1, M1, HID_F, l1p);
    k_zero<<<(HID_F + 255) / 256, 256, 0, stream>>>(l1p + N_NODES * HID_F, HID_F);

    // S2 from l1
    k_gram<<<N_NODES * N_NODES, 32, 0, stream>>>(l1p, HID_F, x);
    k_stats<<<1, 256, 0, stream>>>(x, st);
    k_make_S<<<N_NODES, 64, 0, stream>>>(x, adj, st, S2);

    // l2 = S2 @ reshape(relu(l1 @ W2.T + b2), (300, 63)).T
    k_gemm_relu_wmma<<<dim3(OUT_FP / 16, 4), 32, 0, stream>>>(l1p, W2p, b2p,
                                                              HID_F, OUT_F, M2);
    k_smm<<<OUT_F, 64, 0, stream>>>(S2, M2, OUT_F, l2);
}